// TemporalEncoder_69707319214330
// MI455X (gfx1250) — compile-verified
//
#include <hip/hip_runtime.h>

typedef __attribute__((ext_vector_type(2))) float v2f;
typedef __attribute__((ext_vector_type(4))) float v4f;
typedef __attribute__((ext_vector_type(8))) float v8f;

// ---------------------------------------------------------------------------
// Kernel 1: max over x (8192 elements). One workgroup, LDS tree reduction.
// Writes the scalar max to d_ws[0].
// ---------------------------------------------------------------------------
__global__ void te_max_reduce(const float* __restrict__ x, float* __restrict__ ws, int n) {
    __shared__ float smem[1024];
    const int tid = threadIdx.x;
    float m = -3.4e38f;
    for (int i = tid; i < n; i += 1024) {
        m = fmaxf(m, x[i]);
    }
    smem[tid] = m;
    __syncthreads();
    for (int s = 512; s > 0; s >>= 1) {
        if (tid < s) smem[tid] = fmaxf(smem[tid], smem[tid + s]);
        __syncthreads();
    }
    if (tid == 0) ws[0] = smem[0];
}

// ---------------------------------------------------------------------------
// Kernel 2: streaming zero-fill of rows 1..4095 (the store-bound 134MB part).
// 128-bit nontemporal stores -> global_store_b128 with NT hint.
// ---------------------------------------------------------------------------
__global__ void te_zero_fill(v4f* __restrict__ out, long long n4) {
    long long i = (long long)blockIdx.x * blockDim.x + threadIdx.x;
    const long long stride = (long long)gridDim.x * blockDim.x;
    v4f z = (v4f)0.0f;
    for (; i < n4; i += stride) {
        __builtin_nontemporal_store(z, &out[i]);
    }
}

// ---------------------------------------------------------------------------
// Kernel 3: row 0 = weight[0][j] * (floor(x[j]/max*255) == t) via WMMA.
// Each wave handles one 16-column tile:
//   A (16x4 f32): column K=0 = weight row-0 values, rest 0
//   B (4x16 f32): row    K=0 = mask values (0/1),   rest 0
//   D = A*B  ->  D[m][n] = w[m]*mask[n];  diagonal D[m][m] is the result.
// A/B layout (32-bit 16x4): VGPR0 = K=0 (lanes 0-15) / K=2 (lanes 16-31),
// VGPR1 = K=1 / K=3 -> a.x holds K=0 for lanes 0-15.
// D layout: VGPR v: lanes 0-15 -> (M=v, N=lane); lanes 16-31 -> (M=v+8, N=lane-16).
// Diagonal: m<8 at (vgpr m, lane m); m>=8 at (vgpr m-8, lane m+16).
// ---------------------------------------------------------------------------
__global__ void te_row0_wmma(const float* __restrict__ x,
                             const float* __restrict__ weight,
                             const int*   __restrict__ tptr,
                             const float* __restrict__ ws,
                             float* __restrict__ out, int n) {
    const int lane = threadIdx.x & 31;
    const int wave = (int)((blockIdx.x * blockDim.x + threadIdx.x) >> 5);
    const int base = wave * 16;
    if (base >= n) return;               // uniform per wave

    const float mx = ws[0];
    const int   t  = tptr[0];

    // All 32 lanes load (clamped col) so EXEC stays all-ones into the WMMA.
    const int col = base + (lane & 15);
    const float xv = x[col];
    const float wv = weight[col];        // weight row 0
    const float xn = xv / mx * 255.0f;   // match reference op order
    const int   xq = (int)floorf(xn);
    const float mv = (xq == t) ? 1.0f : 0.0f;

    const bool lo = lane < 16;
    v2f a; a.x = lo ? wv : 0.0f; a.y = 0.0f;   // A[m,0]=w, A[m,1..3]=0
    v2f b; b.x = lo ? mv : 0.0f; b.y = 0.0f;   // B[0,n]=mask, B[1..3,n]=0
    v8f c = {};
    c = __builtin_amdgcn_wmma_f32_16x16x4_f32(
        /*neg_a=*/false, a, /*neg_b=*/false, b,
        /*c_mod=*/(short)0, c, /*reuse_a=*/false, /*reuse_b=*/false);

    // Pull this lane's diagonal element out of the accumulator.
    const int sel = (lane < 8) ? lane : ((lane >= 24) ? (lane - 24) : 0);
    float v;
    switch (sel) {
        case 0: v = c[0]; break;
        case 1: v = c[1]; break;
        case 2: v = c[2]; break;
        case 3: v = c[3]; break;
        case 4: v = c[4]; break;
        case 5: v = c[5]; break;
        case 6: v = c[6]; break;
        default: v = c[7]; break;
    }
    if (lane < 8)        out[base + lane] = v;          // m = lane
    else if (lane >= 24) out[base + lane - 16] = v;     // m = lane - 16
}

// ---------------------------------------------------------------------------
// Launch
// ---------------------------------------------------------------------------
extern "C" void kernel_launch(void* const* d_in, const int* in_sizes, int n_in,
                              void* d_out, int out_size, void* d_ws, size_t ws_size,
                              hipStream_t stream) {
    const float* x      = (const float*)d_in[0];   // (1, 8192) f32
    const float* weight = (const float*)d_in[1];   // (4096, 8192) f32
    const int*   t      = (const int*)d_in[2];     // scalar int
    float* out = (float*)d_out;                    // (4096, 8192) f32
    float* ws  = (float*)d_ws;                     // scratch: ws[0] = max(x)

    const int in_features = in_sizes[0];           // 8192

    // 1) max(x) -> ws[0]
    te_max_reduce<<<1, 1024, 0, stream>>>(x, ws, in_features);

    // 2) zero rows 1..end with 128-bit streaming stores
    const long long tail_floats = (long long)out_size - in_features;
    const long long n4 = tail_floats / 4;
    te_zero_fill<<<8192, 256, 0, stream>>>((v4f*)(out + in_features), n4);

    // 3) row 0 via WMMA outer-product diagonal (one wave per 16 columns)
    const int waves  = in_features / 16;                 // 512
    const int blocks = (waves * 32 + 255) / 256;         // 64 blocks x 8 waves
    te_row0_wmma<<<blocks, 256, 0, stream>>>(x, weight, t, ws, out, in_features);
}